// FakeRouter_62878321214320
// MI455X (gfx1250) — compile-verified
//
#include <hip/hip_runtime.h>

// ---------------------------------------------------------------------------
// MoE router for MI455X (gfx1250).
//   logits = x[T,H] @ weight[E,H]^T + bias ; softmax ; K==E so "full" == scores
// Roofline: 256MB of x @ 23.3TB/s ~= 11us floor; 1.07 GFLOP is negligible, so
// we stay in f32 and use V_WMMA_F32_16X16X4_F32 purely to make the dot
// products free while streaming x once.
// ---------------------------------------------------------------------------

typedef __attribute__((ext_vector_type(2))) float v2f;
typedef __attribute__((ext_vector_type(8))) float v8f;

#define HDIM 4096
#define NEXP 8
#define TOK_PER_TILE 16
#define KSLICES 4            // waves splitting the H reduction per tile
#define TILES_PER_BLOCK 2    // 8 waves per block
#define LDS_STRIDE 17        // 16 experts + 1 pad float to spread LDS banks

__global__ __launch_bounds__(256)
void FakeRouter_wmma_kernel(const float* __restrict__ x,
                            const float* __restrict__ weight,
                            const float* __restrict__ bias,
                            float* __restrict__ out_scores,   // [T, 8]
                            float* __restrict__ out_idx)      // [T, 8] (indices as float)
{
    __shared__ float lds[TILES_PER_BLOCK * KSLICES * TOK_PER_TILE * LDS_STRIDE];

    const int tid  = threadIdx.x;
    const int lane = tid & 31;
    const int wave = tid >> 5;
    const int tileInBlock = wave >> 2;   // 0..1
    const int kslice      = wave & 3;    // 0..3
    const int tile = blockIdx.x * TILES_PER_BLOCK + tileInBlock;
    const int t0   = tile * TOK_PER_TILE;

    const int mrow = lane & 15;          // token-in-tile (A rows) / expert col (B cols)
    const int hi   = lane >> 4;          // lane half selects K sub-pair

    // A: x rows, each lane streams 16B/iter of its token's row (lane-half offset +16B).
    const float* aptr = x + (size_t)(t0 + mrow) * HDIM + kslice * (HDIM / KSLICES) + 4 * hi;
    // B: weight rows; experts 8..15 are masked to zero (weight row e&7 re-read, L0-hit).
    const float* bptr = weight + (size_t)(mrow & 7) * HDIM + kslice * (HDIM / KSLICES) + 4 * hi;
    const float bmask = (mrow < NEXP) ? 1.0f : 0.0f;

    v8f acc = {};

    // Each iteration: two b128 loads feed two 16x16x4 f32 WMMAs (K += 8).
    // Permuted-k mapping: low lane-half supplies logical K{0,1} = phys {k,k+1},
    // high half logical K{2,3} = phys {k+4,k+5} — applied identically to A and B,
    // so the reduction order change is benign.
    #pragma unroll 4
    for (int kk = 0; kk < HDIM / KSLICES; kk += 8) {
        float4 av = *(const float4*)(aptr + kk);
        float4 bv = *(const float4*)(bptr + kk);
        v2f a0 = { av.x, av.y };
        v2f a1 = { av.z, av.w };
        v2f b0 = { bv.x * bmask, bv.y * bmask };
        v2f b1 = { bv.z * bmask, bv.w * bmask };
        acc = __builtin_amdgcn_wmma_f32_16x16x4_f32(false, a0, false, b0,
                                                    (short)0, acc, false, false);
        acc = __builtin_amdgcn_wmma_f32_16x16x4_f32(false, a1, false, b1,
                                                    (short)0, acc, false, false);
    }

    // Stage partial logits: C/D layout => lane holds D[j + 8*hi][mrow] in acc[j].
    float* slice = lds + (tileInBlock * KSLICES + kslice) * (TOK_PER_TILE * LDS_STRIDE);
    #pragma unroll
    for (int j = 0; j < 8; ++j) {
        slice[(j + 8 * hi) * LDS_STRIDE + mrow] = acc[j];
    }

    __syncthreads();

    // Phase 2: 32 tokens per block, one per thread (threads 0..31).
    if (tid < TILES_PER_BLOCK * TOK_PER_TILE) {
        const int tileL = tid >> 4;
        const int tokL  = tid & 15;
        const int t     = blockIdx.x * (TILES_PER_BLOCK * TOK_PER_TILE) + tid;

        float logit[NEXP];
        #pragma unroll
        for (int e = 0; e < NEXP; ++e) logit[e] = bias[e];
        #pragma unroll
        for (int s = 0; s < KSLICES; ++s) {
            const float* sl = lds + (tileL * KSLICES + s) * (TOK_PER_TILE * LDS_STRIDE)
                                  + tokL * LDS_STRIDE;
            #pragma unroll
            for (int e = 0; e < NEXP; ++e) logit[e] += sl[e];
        }

        // softmax over 8
        float mx = logit[0];
        #pragma unroll
        for (int e = 1; e < NEXP; ++e) mx = fmaxf(mx, logit[e]);
        float p[NEXP];
        float sum = 0.0f;
        #pragma unroll
        for (int e = 0; e < NEXP; ++e) { p[e] = __expf(logit[e] - mx); sum += p[e]; }
        const float inv = 1.0f / sum;
        #pragma unroll
        for (int e = 0; e < NEXP; ++e) p[e] *= inv;

        // K == E: scattered "full" output is simply the softmax in original order.
        #pragma unroll
        for (int e = 0; e < NEXP; ++e) out_scores[(size_t)t * NEXP + e] = p[e];

        // Descending arg-ranking, stable (ties -> lower index first, like lax.top_k).
        unsigned used = 0;
        #pragma unroll
        for (int r = 0; r < NEXP; ++r) {
            int   best = 0;
            float bv   = -3.402823466e+38f;
            #pragma unroll
            for (int e = 0; e < NEXP; ++e) {
                if (!((used >> e) & 1u) && p[e] > bv) { bv = p[e]; best = e; }
            }
            used |= 1u << best;
            out_idx[(size_t)t * NEXP + r] = (float)best;
        }
    }
}

extern "C" void kernel_launch(void* const* d_in, const int* in_sizes, int n_in,
                              void* d_out, int out_size, void* d_ws, size_t ws_size,
                              hipStream_t stream) {
    (void)n_in; (void)out_size; (void)d_ws; (void)ws_size;
    const float* x      = (const float*)d_in[0];
    const float* weight = (const float*)d_in[1];
    const float* bias   = (const float*)d_in[2];

    const int T = in_sizes[0] / HDIM;              // 16384 tokens
    float* out_scores = (float*)d_out;             // [T, 8]
    float* out_idx    = (float*)d_out + (size_t)T * NEXP;

    const int tiles  = T / TOK_PER_TILE;           // 1024
    const int blocks = tiles / TILES_PER_BLOCK;    // 512 blocks x 256 thr = 4096 waves

    FakeRouter_wmma_kernel<<<blocks, 256, 0, stream>>>(x, weight, bias,
                                                       out_scores, out_idx);
}